// lagrangian_ode_func_54640573940227
// MI455X (gfx1250) — compile-verified
//
#include <hip/hip_runtime.h>

#define BLOCK 256
#define TILES 8   // rows per thread; 2048 rows per block

typedef float fv4 __attribute__((ext_vector_type(4)));
typedef int   iv4 __attribute__((ext_vector_type(4)));

#if defined(__has_builtin)
#  if __has_builtin(__builtin_amdgcn_global_load_async_to_lds_b128)
#    define HAVE_ASYNC_LDS 1
#  endif
#endif
#ifndef HAVE_ASYNC_LDS
#  define HAVE_ASYNC_LDS 0
#endif

#define NBUF 3  // async pipeline depth

// Exact parameter types per the clang diagnostic: (AS1 iv4*, AS3 iv4*, Imm, Imm)
#define ASYNC_LOAD_B128(gsrc, ldst)                                            \
    __builtin_amdgcn_global_load_async_to_lds_b128(                            \
        (__attribute__((address_space(1))) iv4*)(gsrc),                        \
        (__attribute__((address_space(3))) iv4*)(ldst), 0, 0)

__device__ __forceinline__ fv4 ode_row(fv4 v, float k1, float k2) {
    fv4 o;
    const float d = v.x - v.y;      // q0 - q1
    o.x = v.z;                      // qd0
    o.y = v.w;                      // qd1
    o.z = -(fmaf(k1, v.x, k2 * d)); // qdd0 = -(k1*q0 + k2*(q0-q1))
    o.w = k2 * d;                   // qdd1 =  k2*(q0-q1)
    return o;
}

__global__ __launch_bounds__(BLOCK) void lagrangian_ode_kernel(
    const fv4* __restrict__ x,
    const float* __restrict__ k1p,
    const float* __restrict__ k2p,
    fv4* __restrict__ out,
    int nrows)
{
    const int tid  = threadIdx.x;
    const int base = (int)blockIdx.x * (BLOCK * TILES);
    const float k1 = k1p[0];
    const float k2 = k2p[0];

#if HAVE_ASYNC_LDS
    __shared__ fv4 smem[NBUF][BLOCK];

    // ---- prologue: prime the async pipeline with tiles 0 and 1 ----
    {
        const int r0 = base + 0 * BLOCK + tid;
        if (r0 < nrows) {
            ASYNC_LOAD_B128(x + r0, &smem[0][tid]);
        }
        const int r1 = base + 1 * BLOCK + tid;
        if (r1 < nrows) {
            ASYNC_LOAD_B128(x + r1, &smem[1][tid]);
        }
    }

    // ---- steady state: issue tile t+2, wait for tile t, consume, NT store ----
#pragma unroll
    for (int t = 0; t < TILES; ++t) {
        if (t + 2 < TILES) {
            const int r = base + (t + 2) * BLOCK + tid;
            if (r < nrows) {
                ASYNC_LOAD_B128(x + r, &smem[(t + 2) % NBUF][tid]);
            }
            asm volatile("s_wait_asynccnt 0x2" ::: "memory"); // tiles t+1, t+2 in flight
        } else if (t + 1 < TILES) {
            asm volatile("s_wait_asynccnt 0x1" ::: "memory"); // tile t+1 in flight
        } else {
            asm volatile("s_wait_asynccnt 0x0" ::: "memory"); // drain
        }

        const int r = base + t * BLOCK + tid;
        if (r < nrows) {
            const fv4 v = smem[t % NBUF][tid];                     // ds_load_b128 (own 16B slot)
            __builtin_nontemporal_store(ode_row(v, k1, k2), out + r); // global_store_b128 TH=NT
        }
    }
#else
    // Fallback: direct non-temporal 128-bit streaming (still bandwidth-optimal).
#pragma unroll
    for (int t = 0; t < TILES; ++t) {
        const int r = base + t * BLOCK + tid;
        if (r < nrows) {
            const fv4 v = __builtin_nontemporal_load(x + r);
            __builtin_nontemporal_store(ode_row(v, k1, k2), out + r);
        }
    }
#endif
}

extern "C" void kernel_launch(void* const* d_in, const int* in_sizes, int n_in,
                              void* d_out, int out_size, void* d_ws, size_t ws_size,
                              hipStream_t stream) {
    // setup_inputs order: t (unused), x, k1, k2
    const fv4*   x   = (const fv4*)d_in[1];
    const float* k1p = (const float*)d_in[2];
    const float* k2p = (const float*)d_in[3];
    fv4*         out = (fv4*)d_out;

    const int nrows        = in_sizes[1] / 4;          // B = 8388608
    const int rowsPerBlock = BLOCK * TILES;            // 2048
    const int grid         = (nrows + rowsPerBlock - 1) / rowsPerBlock;

    lagrangian_ode_kernel<<<grid, BLOCK, 0, stream>>>(x, k1p, k2p, out, nrows);
}